// HDPHMM_21440476742013
// MI455X (gfx1250) — compile-verified
//
#include <hip/hip_runtime.h>
#include <math.h>

typedef __attribute__((ext_vector_type(2))) float v2f;
typedef __attribute__((ext_vector_type(8))) float v8f;

#define KSTATES 20
#define FDIM 16
#define EPSF 1e-10f
#define VAR_EPSF 1e-6f
#define LOG2PI 1.8378770664093453f

// workspace layout (float offsets)
#define WS_B 0        // 1024 floats: WMMA B fragments, pre-swizzled per-lane
#define WS_C 1024     // 32 floats : per-state constant (padded)
#define WS_BW 1056    // 32 floats : stick-breaking weights
#define WS_TRANS 1088 // 400 floats: softmax(pi_logits) row-major

__device__ __forceinline__ float rl(float v, int lane) {
  return __builtin_bit_cast(float,
      __builtin_amdgcn_readlane(__builtin_bit_cast(int, v), lane));
}

// wave-uniform sum of x over lanes 0..19 (lanes >=20 must carry 0-contributions)
__device__ __forceinline__ float wave_sum20(float x) {
  float a0 = 0.f, a1 = 0.f, a2 = 0.f, a3 = 0.f;
#pragma unroll
  for (int i = 0; i < KSTATES; i += 4) {
    a0 += rl(x, i);
    a1 += rl(x, i + 1);
    a2 += rl(x, i + 2);
    a3 += rl(x, i + 3);
  }
  return (a0 + a1) + (a2 + a3);
}

// ---------------------------------------------------------------------------
// Prep: stick-breaking weights, softmax(trans), emission constants, and the
// WMMA B matrix W(32,20) pre-swizzled into per-lane fragment layout:
//   A 16x4 f32 frag: lanes 0-15 hold K={0,1}, lanes 16-31 hold K={2,3}
//   B 4x16 f32 frag mirrored: vgpr0/1 -> paired K values, N = lane&15
// ---------------------------------------------------------------------------
__global__ void prep_kernel(const float* __restrict__ beta_logits,
                            const float* __restrict__ pi_logits,
                            const float* __restrict__ means,
                            const float* __restrict__ log_vars,
                            float* __restrict__ ws) {
  __shared__ float inv_s[KSTATES][FDIM];
  __shared__ float w1_s[KSTATES][FDIM];
  __shared__ float cc_s[KSTATES];
  const int lane = threadIdx.x;

  for (int idx = lane; idx < KSTATES * FDIM; idx += 32) {
    int k = idx / FDIM, f = idx % FDIM;
    float var = __expf(log_vars[idx]) + VAR_EPSF;
    float iv = 1.0f / var;
    inv_s[k][f] = iv;
    w1_s[k][f] = -2.0f * means[idx] * iv;
  }
  __syncthreads();

  if (lane < KSTATES) {
    float logdet = 0.f, m2 = 0.f;
    for (int f = 0; f < FDIM; ++f) {
      float iv = inv_s[lane][f];
      logdet += logf(1.0f / iv);
      float m = means[lane * FDIM + f];
      m2 += m * m * iv;
    }
    cc_s[lane] = FDIM * LOG2PI + logdet + m2;
  }
  __syncthreads();

  ws[WS_C + lane] = (lane < KSTATES) ? cc_s[lane] : 0.0f;

  // softmax rows of pi_logits (matches jax.nn.softmax: subtract row max)
  if (lane < KSTATES) {
    float row[KSTATES];
    float mx = -INFINITY;
    for (int j = 0; j < KSTATES; ++j) {
      row[j] = pi_logits[lane * KSTATES + j];
      mx = fmaxf(mx, row[j]);
    }
    float s = 0.f;
    for (int j = 0; j < KSTATES; ++j) {
      row[j] = __expf(row[j] - mx);
      s += row[j];
    }
    float is = 1.0f / s;
    for (int j = 0; j < KSTATES; ++j)
      ws[WS_TRANS + lane * KSTATES + j] = row[j] * is;
  }

  // stick-breaking weights (serial, K=20)
  if (lane == 0) {
    float cum = 1.0f;
    for (int k = 0; k < KSTATES; ++k) {
      float b = 1.0f / (1.0f + __expf(-beta_logits[k]));
      ws[WS_BW + k] = b * cum;
      cum *= (1.0f - b);
    }
  }

  // B fragments: W[f][n], f in [0,32): f<16 -> -2*m*inv pairs with obs,
  //                                    f>=16 -> inv pairs with obs^2
  const int n16 = lane & 15, h = lane >> 4;
  for (int u = 0; u < 16; ++u) {  // u = ct*8 + s
    int ct = u >> 3, s = u & 7;
    int n = ct * 16 + n16;
    int k0 = 4 * s + 2 * h;
    float b0 = 0.f, b1 = 0.f;
    if (n < KSTATES) {
      b0 = (k0 < FDIM) ? w1_s[n][k0] : inv_s[n][k0 - FDIM];
      b1 = (k0 + 1 < FDIM) ? w1_s[n][k0 + 1] : inv_s[n][k0 + 1 - FDIM];
    }
    ws[WS_B + (u * 32 + lane) * 2 + 0] = b0;
    ws[WS_B + (u * 32 + lane) * 2 + 1] = b1;
  }
}

// ---------------------------------------------------------------------------
// Emission: exp_emis(T,K) via V_WMMA_F32_16X16X4_F32 over augmented K=32
// GEMM; result written into BOTH alpha and beta regions of d_out (the scans
// consume them in place).
// ---------------------------------------------------------------------------
#define AUG(i) ((i) < 16 ? f[(i)] : f[(i) - 16] * f[(i) - 16])

__global__ __launch_bounds__(256) void emis_kernel(
    const float* __restrict__ obs, const float* __restrict__ ws,
    float* __restrict__ outA, float* __restrict__ outB, int nTiles) {
  const int lane = threadIdx.x & 31;
  const int wave = (blockIdx.x * blockDim.x + threadIdx.x) >> 5;
  const int nWaves = (gridDim.x * blockDim.x) >> 5;
  const int n16 = lane & 15, h = lane >> 4;

  // hoist B fragments + state constants (wave-invariant)
  v2f bfr[16];
#pragma unroll
  for (int u = 0; u < 16; ++u) {
    bfr[u].x = ws[WS_B + (u * 32 + lane) * 2 + 0];
    bfr[u].y = ws[WS_B + (u * 32 + lane) * 2 + 1];
  }
  const float cc0 = ws[WS_C + n16];
  const float cc1 = ws[WS_C + 16 + n16];

  for (int tile = wave; tile < nTiles; tile += nWaves) {
    const float4* rowp =
        (const float4*)(obs + (size_t)(tile * 16 + n16) * FDIM);
    float4 q0 = rowp[0], q1 = rowp[1], q2 = rowp[2], q3 = rowp[3];
    float f[16] = {q0.x, q0.y, q0.z, q0.w, q1.x, q1.y, q1.z, q1.w,
                   q2.x, q2.y, q2.z, q2.w, q3.x, q3.y, q3.z, q3.w};

    v8f c0 = {0.f, 0.f, 0.f, 0.f, 0.f, 0.f, 0.f, 0.f};
    v8f c1 = {0.f, 0.f, 0.f, 0.f, 0.f, 0.f, 0.f, 0.f};
#pragma unroll
    for (int s = 0; s < 8; ++s) {
      const int base = 4 * s;
      float ax_lo = AUG(base), ay_lo = AUG(base + 1);
      float ax_hi = AUG(base + 2), ay_hi = AUG(base + 3);
      v2f a;
      a.x = h ? ax_hi : ax_lo;
      a.y = h ? ay_hi : ay_lo;
      c0 = __builtin_amdgcn_wmma_f32_16x16x4_f32(false, a, false, bfr[s],
                                                 (short)0, c0, false, false);
      c1 = __builtin_amdgcn_wmma_f32_16x16x4_f32(false, a, false, bfr[8 + s],
                                                 (short)0, c1, false, false);
    }

#pragma unroll
    for (int v = 0; v < 8; ++v) {
      size_t t = (size_t)tile * 16 + h * 8 + v;  // C/D layout: M = v + 8*h
      float e0 = __expf(-0.5f * (c0[v] + cc0));
      outA[t * KSTATES + n16] = e0;
      outB[t * KSTATES + n16] = e0;
      if (n16 < KSTATES - 16) {  // states 16..19 only
        float e1 = __expf(-0.5f * (c1[v] + cc1));
        outA[t * KSTATES + 16 + n16] = e1;
        outB[t * KSTATES + 16 + n16] = e1;
      }
    }
  }
}

// ---------------------------------------------------------------------------
// Serial scans: block 0 = forward (consumes/overwrites alpha region in place),
// block 1 = backward (consumes/overwrites beta region in place). One wave
// each; readlane broadcast + 8-deep e-prefetch ring.
// ---------------------------------------------------------------------------
__global__ __launch_bounds__(32) void scan_kernel(const float* __restrict__ ws,
                                                  float* outA, float* outB,
                                                  float* ll, int T) {
  const int lane = threadIdx.x;
  const float* trans = ws + WS_TRANS;

  if (blockIdx.x == 0) {
    // ---- forward: a_t = normalize((a_{t-1} @ trans) * e_t) ----
    float tcol[KSTATES];
#pragma unroll
    for (int i = 0; i < KSTATES; ++i)
      tcol[i] = (lane < KSTATES) ? trans[i * KSTATES + lane] : 0.0f;
    float bw = (lane < KSTATES) ? ws[WS_BW + lane] : 0.0f;

    float e0 = (lane < KSTATES) ? outA[lane] : 0.0f;
    float x = bw * e0;
    float s = wave_sum20(x);
    float r = __builtin_amdgcn_rcpf(s + EPSF);
    float a = x * r;
    if (lane < KSTATES) outA[lane] = a;

    float ebuf[8];
#pragma unroll
    for (int d = 1; d <= 8; ++d)
      ebuf[d & 7] = (lane < KSTATES && d < T) ? outA[(size_t)d * KSTATES + lane]
                                              : 0.0f;

    for (int t = 1; t < T; ++t) {
      float e = ebuf[t & 7];
      int tp = t + 8;
      ebuf[t & 7] = (lane < KSTATES && tp < T)
                        ? outA[(size_t)tp * KSTATES + lane]
                        : 0.0f;
      float a0 = 0.f, a1 = 0.f, a2 = 0.f, a3 = 0.f;
#pragma unroll
      for (int i = 0; i < KSTATES; i += 4) {
        a0 += rl(a, i) * tcol[i];
        a1 += rl(a, i + 1) * tcol[i + 1];
        a2 += rl(a, i + 2) * tcol[i + 2];
        a3 += rl(a, i + 3) * tcol[i + 3];
      }
      x = ((a0 + a1) + (a2 + a3)) * e;
      s = wave_sum20(x);
      r = __builtin_amdgcn_rcpf(s + EPSF);
      a = x * r;
      if (lane < KSTATES) outA[(size_t)t * KSTATES + lane] = a;
    }
    // log_likelihood = log(sum(alpha[-1]) + EPS); sum(alpha[-1]) = s*r
    if (lane == 0) *ll = logf(s * r + EPSF);
  } else {
    // ---- backward: b_t = normalize(trans @ (b_{t+1} * e_{t+1})) ----
    float trow[KSTATES];
#pragma unroll
    for (int j = 0; j < KSTATES; ++j)
      trow[j] = (lane < KSTATES) ? trans[lane * KSTATES + j] : 0.0f;
    float b = (lane < KSTATES) ? 1.0f : 0.0f;

    float ebuf[8];
#pragma unroll
    for (int d = 0; d < 8; ++d) {
      int tt = T - 1 - d;
      ebuf[tt & 7] = (lane < KSTATES && tt >= 1)
                         ? outB[(size_t)tt * KSTATES + lane]
                         : 0.0f;
    }

    for (int t = T - 2; t >= 0; --t) {
      int tt = t + 1;
      float e = ebuf[tt & 7];
      int tpre = tt - 8;
      ebuf[tt & 7] = (lane < KSTATES && tpre >= 1)
                         ? outB[(size_t)tpre * KSTATES + lane]
                         : 0.0f;
      float w = b * e;
      float a0 = 0.f, a1 = 0.f, a2 = 0.f, a3 = 0.f;
#pragma unroll
      for (int j = 0; j < KSTATES; j += 4) {
        a0 += rl(w, j) * trow[j];
        a1 += rl(w, j + 1) * trow[j + 1];
        a2 += rl(w, j + 2) * trow[j + 2];
        a3 += rl(w, j + 3) * trow[j + 3];
      }
      float x = (a0 + a1) + (a2 + a3);
      float s = wave_sum20(x);
      float r = __builtin_amdgcn_rcpf(s + EPSF);
      b = x * r;
      if (lane < KSTATES) outB[(size_t)t * KSTATES + lane] = b;
    }
    // beta[T-1] = ones, written AFTER its e slot was consumed
    if (lane < KSTATES) outB[(size_t)(T - 1) * KSTATES + lane] = 1.0f;
  }
}

extern "C" void kernel_launch(void* const* d_in, const int* in_sizes, int n_in,
                              void* d_out, int out_size, void* d_ws,
                              size_t ws_size, hipStream_t stream) {
  const float* obs = (const float*)d_in[0];
  const float* beta_logits = (const float*)d_in[1];
  const float* pi_logits = (const float*)d_in[2];
  const float* means = (const float*)d_in[3];
  const float* log_vars = (const float*)d_in[4];

  const int T = in_sizes[0] / FDIM;  // 1048576
  float* out = (float*)d_out;
  float* outA = out;                            // alpha (T,K)
  float* outB = out + (size_t)T * KSTATES;      // beta  (T,K)
  float* ll = out + 2 * (size_t)T * KSTATES;    // scalar
  float* ws = (float*)d_ws;

  prep_kernel<<<1, 32, 0, stream>>>(beta_logits, pi_logits, means, log_vars,
                                    ws);
  const int nTiles = T / 16;
  emis_kernel<<<1024, 256, 0, stream>>>(obs, ws, outA, outB, nTiles);
  scan_kernel<<<2, 32, 0, stream>>>(ws, outA, outB, ll, T);
}